// DirectionalConv2D_39384850104875
// MI455X (gfx1250) — compile-verified
//
#include <hip/hip_runtime.h>
#include <stdint.h>

// DirectionalConv2D on gfx1250 (MI455X):
//   per-pixel anisotropic 5x5 Gaussian from wind direction, normalized,
//   convolved over zero-padded fire map, residual blend + clamp.
//
// Strategy: 16x16 output tile / 256-thread block (8 wave32), 20x20 input tile
// staged into LDS via CDNA5 async global->LDS DMA (ASYNCcnt path). Weights
// factorized as A^(i^2) * B^(j^2) * C^(i*j) -> 3 exps + 1 rsq + 1 rcp per
// pixel instead of 25 exps + atan2/cos/sin.

#define KW    5
#define PAD   2
#define TILE  16
#define LDSW  (TILE + 2 * PAD)   // 20
#define LDSN  (LDSW * LDSW)      // 400

__global__ __launch_bounds__(256)
void dirconv5x5_kernel(const float* __restrict__ fire,
                       const float* __restrict__ wu,
                       const float* __restrict__ wv,
                       float* __restrict__ out,
                       int H, int W)
{
    __shared__ float smem[LDSN];

    const int tx = threadIdx.x;            // 0..15
    const int ty = threadIdx.y;            // 0..15
    const int t  = ty * TILE + tx;         // 0..255
    const int bx = blockIdx.x * TILE;
    const int by = blockIdx.y * TILE;
    const int b  = blockIdx.z;

    const float* fireB = fire + (size_t)b * H * W;   // uniform -> SGPR pair

    // ---- 1) zero-init the 20x20 LDS tile: provides the zero-padding halo ----
    smem[t] = 0.0f;
    if (t + 256 < LDSN) smem[t + 256] = 0.0f;
    // DS and ASYNC counters are mutually unordered; each LDS address is zeroed
    // and DMA-written by the same thread, so draining DScnt before issuing the
    // async DMA gives correct per-address ordering.
    asm volatile("s_wait_dscnt 0x0" ::: "memory");

    // ---- 2) async global->LDS DMA of in-bounds tile elements (CDNA5) ----
    #pragma unroll
    for (int ph = 0; ph < 2; ++ph) {
        const int idx = t + ph * 256;
        if (idx < LDSN) {
            const int ly = idx / LDSW;
            const int lx = idx - ly * LDSW;
            const int gy = by + ly - PAD;
            const int gx = bx + lx - PAD;
            if (gy >= 0 && gy < H && gx >= 0 && gx < W) {
                const uint32_t lds_off = (uint32_t)(uintptr_t)(&smem[idx]); // LDS byte offset in addr[31:0]
                const int      goff    = (gy * W + gx) * 4;                 // byte offset from SGPR base
                asm volatile(
                    "global_load_async_to_lds_b32 %0, %1, %2 offset:0"
                    :
                    : "v"(lds_off), "v"(goff), "s"(fireB)
                    : "memory");
            }
        }
    }
    asm volatile("s_wait_asynccnt 0x0" ::: "memory");
    __syncthreads();

    // ---- 3) per-pixel factorized directional Gaussian + stencil ----
    const int y = by + ty;
    const int x = bx + tx;
    const size_t p = (size_t)b * H * W + (size_t)y * W + x;

    const float kk = 1.0f / (2.0f * 1.5f * 1.5f);  // 1/(2*sigma^2)

    // cos/sin(atan2(v, u+1e-8)) == normalized direction vector
    const float u  = wu[p] + 1e-8f;
    const float v  = wv[p];
    const float r2 = fmaxf(u * u + v * v, 1e-38f);
    const float ri = __builtin_amdgcn_rsqf(r2);
    const float c  = u * ri;
    const float s  = v * ri;

    // weight(i,j) = exp(-((i*c + j*s)^2)*kk) = A^(i^2) * B^(j^2) * C^(i*j)
    const float A   = __expf(-(c * c) * kk);
    const float Bv  = __expf(-(s * s) * kk);
    const float Cm  = __expf(-(2.0f * c * s) * kk);   // |exponent| <= 0.223, safe
    const float A2  = A * A,   A4 = A2 * A2;
    const float B2  = Bv * Bv, B4 = B2 * B2;
    const float Cp2 = Cm * Cm, Cp4 = Cp2 * Cp2;
    const float Cn1 = __builtin_amdgcn_rcpf(Cm);      // Cm in [0.80, 1.25]
    const float Cn2 = Cn1 * Cn1, Cn4 = Cn2 * Cn2;

    const float ai[5] = { A4, A, 1.0f, A, A4 };                 // index i+2 (i pairs with cos / dy)
    const float bj[5] = { B4, Bv, 1.0f, Bv, B4 };               // index j+2 (j pairs with sin / dx)
    const float cp[9] = { Cn4, 0.0f, Cn2, Cn1, 1.0f, Cm, Cp2, 0.0f, Cp4 }; // index i*j+4 (|ij|==3 unused)

    float acc = 0.0f, wsum = 0.0f;
    #pragma unroll
    for (int i = -PAD; i <= PAD; ++i) {
        const float wi = ai[i + PAD];
        #pragma unroll
        for (int j = -PAD; j <= PAD; ++j) {
            const float w    = wi * bj[j + PAD] * cp[i * j + 4];
            const float pval = smem[(ty + PAD + i) * LDSW + (tx + PAD + j)];
            wsum += w;
            acc   = fmaf(w, pval, acc);
        }
    }

    const float center = smem[(ty + PAD) * LDSW + (tx + PAD)];  // == fire[p]
    const float spread = acc * __builtin_amdgcn_rcpf(wsum + 1e-8f);
    float res = 0.7f * spread + 0.3f * center;
    res = fminf(fmaxf(res, 0.0f), 1.0f);
    out[p] = res;
}

extern "C" void kernel_launch(void* const* d_in, const int* in_sizes, int n_in,
                              void* d_out, int out_size, void* d_ws, size_t ws_size,
                              hipStream_t stream) {
    const float* fire = (const float*)d_in[0];
    const float* wu   = (const float*)d_in[1];
    const float* wv   = (const float*)d_in[2];
    float*       out  = (float*)d_out;

    const int H = 512, W = 512;
    const int B = in_sizes[0] / (H * W);     // 4

    dim3 grid(W / TILE, H / TILE, B);
    dim3 block(TILE, TILE, 1);
    hipLaunchKernelGGL(dirconv5x5_kernel, grid, block, 0, stream,
                       fire, wu, wv, out, H, W);
}